// LigerFusedLinearCrossEntropyLoss_7421703487602
// MI455X (gfx1250) — compile-verified
//
#include <hip/hip_runtime.h>
#include <hip/hip_bf16.h>

typedef __attribute__((ext_vector_type(8))) float v8f;
typedef __attribute__((ext_vector_type(2))) float v2f;
typedef __attribute__((ext_vector_type(4))) int   v4i;

#define BM 64
#define BN 128
#define BK 32
#define LDAP (BK + 4)   // 36-float row stride: 16B-aligned, conflict-free frag reads
#define IGNORE_INDEX (-100)
#define NSPLIT 10

#if defined(__gfx1250__) && __has_builtin(__builtin_amdgcn_global_load_async_to_lds_b128)
#define HAS_ASYNC_LDS 1
#else
#define HAS_ASYNC_LDS 0
#endif

#if HAS_ASYNC_LDS
typedef __attribute__((address_space(1))) v4i v4i_glob;
typedef __attribute__((address_space(3))) v4i v4i_lds;
#endif

__device__ __forceinline__ void async_cp16(const float* g, float* l) {
#if HAS_ASYNC_LDS
    __builtin_amdgcn_global_load_async_to_lds_b128(
        (v4i_glob*)g, (v4i_lds*)l, 0, 0);
#else
    *(float4*)l = *(const float4*)g;
#endif
}

__device__ __forceinline__ void wait_async_lds() {
#if HAS_ASYNC_LDS
#if __has_builtin(__builtin_amdgcn_s_wait_asynccnt)
    __builtin_amdgcn_s_wait_asynccnt(0);
#else
    asm volatile("s_wait_asynccnt 0" ::: "memory");
#endif
#endif
}

// Kernel 1: fused GEMM (X @ W^T + bias) with online softmax statistics.
// Each block: 64 rows x one vocab split (V/NSPLIT columns), tiled 128 cols at a time.
// Software pipeline: async DMA of tile (s+1) into the alternate LDS buffer while
// WMMAs consume tile s; one barrier per K-step.
__global__ __launch_bounds__(256)
void flce_gemm_kernel(const float* __restrict__ W,     // [V,H]
                      const float* __restrict__ X,     // [N,H]
                      const int*   __restrict__ tgt,   // [N]
                      const float* __restrict__ bias,  // [V]
                      float* __restrict__ m_part,      // [N,NSPLIT]
                      float* __restrict__ s_part,      // [N,NSPLIT]
                      float* __restrict__ t_logit,     // [N]
                      int N, int V, int H)
{
    __shared__ __align__(16) float As[2][BM][LDAP];
    __shared__ __align__(16) float Bs[2][BN][LDAP];
    __shared__ float msh[2][BM];
    __shared__ float ssh[2][BM];
    __shared__ int   tl[BM];

    const int tid  = threadIdx.x;
    const int lane = tid & 31;
    const int wave = tid >> 5;     // 0..7
    const int wr   = wave >> 1;    // wave row 0..3  (16 rows each)
    const int wc   = wave & 1;     // wave col 0..1  (64 cols each)
    const int half = lane >> 4;    // 0: rows +0..7, 1: rows +8..15 (C layout)
    const int l15  = lane & 15;

    const int r0      = blockIdx.x * BM;
    const int split   = blockIdx.y;
    const int colsps  = V / NSPLIT;            // columns per split
    const int v_begin = split * colsps;
    const int ntiles  = colsps / BN;
    const int ksteps  = H / BK;
    const int total   = ntiles * ksteps;

    if (tid < BM) {
        int t = tgt[r0 + tid];
        tl[tid] = t < 0 ? 0 : (t >= V ? V - 1 : t);   // clipped target (reference clips)
    }
    __syncthreads();

    // per-thread loader geometry (6 x 16B per K-step)
    const int lrow = tid >> 3;
    const int lc4  = (tid & 7) << 2;

    auto issue_tiles = [&](int vt, int k0, int buf) {
        const int v0 = v_begin + vt * BN;
        async_cp16(&X[(long)(r0 + lrow) * H + k0 + lc4],      &As[buf][lrow][lc4]);
        async_cp16(&X[(long)(r0 + lrow + 32) * H + k0 + lc4], &As[buf][lrow + 32][lc4]);
#pragma unroll
        for (int q = 0; q < 4; ++q)
            async_cp16(&W[(long)(v0 + lrow + q * 32) * H + k0 + lc4],
                       &Bs[buf][lrow + q * 32][lc4]);
    };

    float m_run[8], s_run[8];
#pragma unroll
    for (int i = 0; i < 8; ++i) { m_run[i] = -__builtin_inff(); s_run[i] = 0.f; }

    const v8f vzero = {};
    v8f acc[4];
#pragma unroll
    for (int j = 0; j < 4; ++j) acc[j] = vzero;

    int it = 0, k0 = 0;
    issue_tiles(0, 0, 0);

    for (int s = 0; s < total; ++s) {
        const int buf = s & 1;

        wait_async_lds();     // my DMA into buf complete
        __syncthreads();      // everyone's DMA complete; buf^1 free for reuse

        if (s + 1 < total) {
            int nk0 = k0 + BK, nit = it;
            if (nk0 == H) { nk0 = 0; ++nit; }
            issue_tiles(nit, nk0, buf ^ 1);   // overlap with compute below
        }

        // ---- 8 x WMMA K-substeps of 4 ----
#pragma unroll
        for (int kk = 0; kk < BK; kk += 4) {
            v2f a;
            a.x = As[buf][wr * 16 + l15][kk + half * 2];
            a.y = As[buf][wr * 16 + l15][kk + half * 2 + 1];
#pragma unroll
            for (int j = 0; j < 4; ++j) {
                v2f b;
                const int col = wc * 64 + j * 16 + l15;
                b.x = Bs[buf][col][kk + half * 2];
                b.y = Bs[buf][col][kk + half * 2 + 1];
                acc[j] = __builtin_amdgcn_wmma_f32_16x16x4_f32(
                    false, a, false, b, (short)0, acc[j], false, false);
            }
        }

        k0 += BK;
        if (k0 == H) {
            // ---- epilogue for this vocab tile: bias, target, online (m,s) ----
            const int v0 = v_begin + it * BN;
            float bcol[4];
#pragma unroll
            for (int j = 0; j < 4; ++j)
                bcol[j] = bias[v0 + wc * 64 + j * 16 + l15];
#pragma unroll
            for (int j = 0; j < 4; ++j)
#pragma unroll
                for (int i = 0; i < 8; ++i)
                    acc[j][i] += bcol[j];

            const int cbase = v0 + wc * 64;
#pragma unroll
            for (int i = 0; i < 8; ++i) {
                const int rl  = wr * 16 + half * 8 + i;
                const int off = tl[rl] - cbase;
                if (off >= 0 && off < 64 && (off & 15) == l15) {
                    const int sel = off >> 4;
                    float cv = acc[0][i];
                    cv = (sel == 1) ? acc[1][i] : cv;
                    cv = (sel == 2) ? acc[2][i] : cv;
                    cv = (sel == 3) ? acc[3][i] : cv;
                    t_logit[r0 + rl] = cv;
                }
            }

#pragma unroll
            for (int i = 0; i < 8; ++i) {
                float v = fmaxf(fmaxf(acc[0][i], acc[1][i]),
                                fmaxf(acc[2][i], acc[3][i]));
#pragma unroll
                for (int m = 8; m >= 1; m >>= 1)
                    v = fmaxf(v, __shfl_xor(v, m, 32));   // reduce across 16-lane half
                const float nm = fmaxf(m_run[i], v);
                float p = __expf(acc[0][i] - nm) + __expf(acc[1][i] - nm) +
                          __expf(acc[2][i] - nm) + __expf(acc[3][i] - nm);
#pragma unroll
                for (int m = 8; m >= 1; m >>= 1)
                    p += __shfl_xor(p, m, 32);
                s_run[i] = s_run[i] * __expf(m_run[i] - nm) + p;
                m_run[i] = nm;
            }

            k0 = 0; ++it;
#pragma unroll
            for (int j = 0; j < 4; ++j) acc[j] = vzero;
        }
    }

    // ---- merge the two wave-columns, write per-(row,split) partials ----
    if (l15 == 0) {
#pragma unroll
        for (int i = 0; i < 8; ++i) {
            const int rl = wr * 16 + half * 8 + i;
            msh[wc][rl] = m_run[i];
            ssh[wc][rl] = s_run[i];
        }
    }
    __syncthreads();
    if (tid < BM) {
        const float m0 = msh[0][tid], m1 = msh[1][tid];
        const float s0 = ssh[0][tid], s1 = ssh[1][tid];
        const float mm = fmaxf(m0, m1);
        const float ss = s0 * __expf(m0 - mm) + s1 * __expf(m1 - mm);
        m_part[(long)(r0 + tid) * NSPLIT + split] = mm;
        s_part[(long)(r0 + tid) * NSPLIT + split] = ss;
    }
}

// Kernel 2: merge split partials per row, masked mean of NLL.
__global__ __launch_bounds__(256)
void flce_merge_kernel(const float* __restrict__ m_part,
                       const float* __restrict__ s_part,
                       const float* __restrict__ t_logit,
                       const int*   __restrict__ tgt,
                       float* __restrict__ out, int N)
{
    __shared__ float sred[256];
    __shared__ float cred[256];
    float lsum = 0.f, lcnt = 0.f;
    for (int n = threadIdx.x; n < N; n += 256) {
        float m = -__builtin_inff();
        for (int s = 0; s < NSPLIT; ++s)
            m = fmaxf(m, m_part[(long)n * NSPLIT + s]);
        float ss = 0.f;
        for (int s = 0; s < NSPLIT; ++s)
            ss += s_part[(long)n * NSPLIT + s] *
                  __expf(m_part[(long)n * NSPLIT + s] - m);
        const float lse = m + __logf(ss);
        const float nll = lse - t_logit[n];
        if (tgt[n] != IGNORE_INDEX) { lsum += nll; lcnt += 1.f; }
    }
    sred[threadIdx.x] = lsum;
    cred[threadIdx.x] = lcnt;
    __syncthreads();
    for (int off = 128; off > 0; off >>= 1) {
        if (threadIdx.x < off) {
            sred[threadIdx.x] += sred[threadIdx.x + off];
            cred[threadIdx.x] += cred[threadIdx.x + off];
        }
        __syncthreads();
    }
    if (threadIdx.x == 0)
        out[0] = sred[0] / fmaxf(cred[0], 1.f);
}

extern "C" void kernel_launch(void* const* d_in, const int* in_sizes, int n_in,
                              void* d_out, int out_size, void* d_ws, size_t ws_size,
                              hipStream_t stream) {
    const float* W    = (const float*)d_in[0];   // [V,H]
    const float* X    = (const float*)d_in[1];   // [N,H]
    const int*   tgt  = (const int*)d_in[2];     // [N]
    const float* bias = (const float*)d_in[3];   // [V]

    const int N = in_sizes[2];
    const int V = in_sizes[3];
    const int H = in_sizes[1] / N;

    float* ws      = (float*)d_ws;
    float* m_part  = ws;                          // N * NSPLIT
    float* s_part  = m_part + (size_t)N * NSPLIT; // N * NSPLIT
    float* t_logit = s_part + (size_t)N * NSPLIT; // N

    dim3 grid(N / BM, NSPLIT);   // row-tile fast-varying => weight split shared in L2
    flce_gemm_kernel<<<grid, 256, 0, stream>>>(W, X, tgt, bias,
                                               m_part, s_part, t_logit, N, V, H);
    flce_merge_kernel<<<1, 256, 0, stream>>>(m_part, s_part, t_logit, tgt,
                                             (float*)d_out, N);
}